// CriticNetwork_8031588844234
// MI455X (gfx1250) — compile-verified
//
#include <hip/hip_runtime.h>

typedef __attribute__((ext_vector_type(2))) float v2f;
typedef __attribute__((ext_vector_type(8))) float v8f;

#define NODES       80000
#define INDIM       128
#define FDIM1       32
#define FDIM2       64
#define NODES_PER_B 10000
#define NBATCH      8

// ---------------- degree / normalization ----------------

__global__ void k_init_deg(float* __restrict__ deg) {
    int n = blockIdx.x * blockDim.x + threadIdx.x;
    if (n < NODES) deg[n] = 1.0f;   // self-loop contributes 1 to every node
}

__global__ void k_degree(const long long* __restrict__ dst, int E,
                         float* __restrict__ deg) {
    int e = blockIdx.x * blockDim.x + threadIdx.x;
    if (e < E) atomicAdd(&deg[(int)dst[e]], 1.0f);
}

__global__ void k_dinv(float* __restrict__ deg) {
    int n = blockIdx.x * blockDim.x + threadIdx.x;
    if (n < NODES) deg[n] = rsqrtf(deg[n]);   // deg >= 1 always
}

// ---------------- dense GEMM via WMMA f32 16x16x4 ----------------
// H[M,N] = X[M,K] @ W[K,N];  M = 16 * gridDim.x, one wave per 16-wide N tile.
// A frag: lane L<16 holds A[M=L][K=k0..k0+1], lanes 16-31 hold K=k0+2..k0+3.
// B frag: lane L<16 holds B[K=k0..k0+1][N=L], lanes 16-31 hold K=k0+2..k0+3.
template <int K, int N>
__global__ void k_gemm_wmma(const float* __restrict__ X,
                            const float* __restrict__ W,
                            float* __restrict__ H) {
    const int row0  = blockIdx.x * 16;
    const int wave  = threadIdx.x >> 5;       // N tile index
    const int lane  = threadIdx.x & 31;
    const int m     = lane & 15;
    const int khalf = (lane >> 4) * 2;        // 0 for lanes 0-15, 2 for 16-31
    const int ncol  = wave * 16 + m;

    v8f acc = {};
#pragma unroll
    for (int kk = 0; kk < K; kk += 4) {
        const int k0 = kk + khalf;
        v2f a, b;
        const float* xp = X + (size_t)(row0 + m) * K + k0;
        a.x = xp[0];
        a.y = xp[1];
        b.x = W[(size_t)(k0 + 0) * N + ncol];
        b.y = W[(size_t)(k0 + 1) * N + ncol];
        acc = __builtin_amdgcn_wmma_f32_16x16x4_f32(
            /*neg_a=*/false, a, /*neg_b=*/false, b,
            /*c_mod=*/(short)0, acc, /*reuse_a=*/false, /*reuse_b=*/false);
    }

    // C/D layout: VGPR j -> M = j (lanes 0-15) / j+8 (lanes 16-31), N = lane&15
    const int rbase = row0 + (lane >> 4) * 8;
#pragma unroll
    for (int j = 0; j < 8; ++j)
        H[(size_t)(rbase + j) * N + ncol] = acc[j];
}

// ---------------- edge gather/scale/scatter-add ----------------
// F features, C features per thread (C multiple of 4), F/C threads per edge.
template <int F, int C>
__global__ void k_edge_scatter(const long long* __restrict__ src,
                               const long long* __restrict__ dst,
                               const float* __restrict__ dinv,
                               const float* __restrict__ H,
                               float* __restrict__ AGG, int E) {
    const int TPE = F / C;
    long long tid = (long long)blockIdx.x * blockDim.x + threadIdx.x;
    long long e = tid / TPE;
    if (e >= E) return;
    const int f0 = (int)(tid % TPE) * C;
    const int s = (int)src[e];
    const int d = (int)dst[e];
    const float norm = dinv[s] * dinv[d];
    const float* hp = H + (size_t)s * F + f0;
    float* ap = AGG + (size_t)d * F + f0;
#pragma unroll
    for (int j = 0; j < C; j += 4) {
        float4 v = *(const float4*)(hp + j);
        atomicAdd(ap + j + 0, v.x * norm);
        atomicAdd(ap + j + 1, v.y * norm);
        atomicAdd(ap + j + 2, v.z * norm);
        atomicAdd(ap + j + 3, v.w * norm);
    }
}

// ---------------- self-loop + bias + relu (in place on AGG) ----------------
template <int F>
__global__ void k_finalize(const float* __restrict__ H,
                           const float* __restrict__ dinv,
                           const float* __restrict__ bias,
                           float* __restrict__ AGG) {
    long long t = (long long)blockIdx.x * blockDim.x + threadIdx.x;
    if (t >= (long long)NODES * F) return;
    const int n = (int)(t / F);
    const int f = (int)(t % F);
    const float di = dinv[n];
    float v = AGG[t] + H[t] * di * di + bias[f];
    AGG[t] = v > 0.0f ? v : 0.0f;
}

// ---------------- per-batch readout: dot(flat[b], Wout) + bout ----------------
__global__ void k_readout(const float* __restrict__ H2,
                          const float* __restrict__ Wout,
                          const float* __restrict__ bout,
                          float* __restrict__ out) {
    __shared__ float red[512];
    const int b = blockIdx.x;
    const int LEN = NODES_PER_B * FDIM2;   // 640000
    const float* base = H2 + (size_t)b * LEN;
    float s = 0.0f;
    for (int i = threadIdx.x; i < LEN; i += blockDim.x)
        s += base[i] * Wout[i];
    red[threadIdx.x] = s;
    __syncthreads();
    for (int w = blockDim.x >> 1; w > 0; w >>= 1) {
        if ((int)threadIdx.x < w) red[threadIdx.x] += red[threadIdx.x + w];
        __syncthreads();
    }
    if (threadIdx.x == 0) out[b] = red[0] + bout[0];
}

// ---------------- launcher ----------------

extern "C" void kernel_launch(void* const* d_in, const int* in_sizes, int n_in,
                              void* d_out, int out_size, void* d_ws, size_t ws_size,
                              hipStream_t stream) {
    const float*     x    = (const float*)d_in[0];
    const long long* ei   = (const long long*)d_in[1];
    const float*     W1   = (const float*)d_in[2];
    const float*     b1   = (const float*)d_in[3];
    const float*     W2   = (const float*)d_in[4];
    const float*     b2   = (const float*)d_in[5];
    const float*     Wout = (const float*)d_in[6];
    const float*     bout = (const float*)d_in[7];
    float*           out  = (float*)d_out;

    const int E = in_sizes[1] / 2;           // 2,560,000
    const long long* src = ei;               // ei[0, :]
    const long long* dst = ei + E;           // ei[1, :]

    float* ws   = (float*)d_ws;
    float* dinv = ws;                                        //   80,000
    float* h1   = dinv + NODES;                              // 2,560,000
    float* agg1 = h1   + (size_t)NODES * FDIM1;              // 2,560,000
    float* h2   = agg1 + (size_t)NODES * FDIM1;              // 5,120,000
    float* agg2 = h2   + (size_t)NODES * FDIM2;              // 5,120,000

    hipMemsetAsync(agg1, 0, (size_t)NODES * FDIM1 * sizeof(float), stream);
    hipMemsetAsync(agg2, 0, (size_t)NODES * FDIM2 * sizeof(float), stream);

    // normalization
    k_init_deg<<<(NODES + 255) / 256, 256, 0, stream>>>(dinv);
    k_degree<<<(E + 255) / 256, 256, 0, stream>>>(dst, E, dinv);
    k_dinv<<<(NODES + 255) / 256, 256, 0, stream>>>(dinv);

    // layer 1: h1 = x @ W1  (WMMA), scatter, finalize(relu) in-place on agg1
    k_gemm_wmma<INDIM, FDIM1><<<NODES / 16, 64, 0, stream>>>(x, W1, h1);
    {
        long long T = (long long)E * (FDIM1 / 16);           // C=16 -> 2 thr/edge
        k_edge_scatter<FDIM1, 16><<<(int)((T + 255) / 256), 256, 0, stream>>>(
            src, dst, dinv, h1, agg1, E);
    }
    k_finalize<FDIM1><<<(int)(((long long)NODES * FDIM1 + 255) / 256), 256, 0, stream>>>(
        h1, dinv, b1, agg1);

    // layer 2: h2 = relu1 @ W2  (WMMA), scatter, finalize(relu) in-place on agg2
    k_gemm_wmma<FDIM1, FDIM2><<<NODES / 16, 128, 0, stream>>>(agg1, W2, h2);
    {
        long long T = (long long)E * (FDIM2 / 16);           // C=16 -> 4 thr/edge
        k_edge_scatter<FDIM2, 16><<<(int)((T + 255) / 256), 256, 0, stream>>>(
            src, dst, dinv, h2, agg2, E);
    }
    k_finalize<FDIM2><<<(int)(((long long)NODES * FDIM2 + 255) / 256), 256, 0, stream>>>(
        h2, dinv, b2, agg2);

    // readout: 8 dot products
    k_readout<<<NBATCH, 512, 0, stream>>>(agg2, Wout, bout, out);
}